// net_30253749633665
// MI455X (gfx1250) — compile-verified
//
#include <hip/hip_runtime.h>
#include <stdint.h>

#define H_LEN 50
#define H_PAD 52                // K padded to multiple of 4 for 16x16x4 WMMA
#define DIM   64
#define WPB   4                 // waves per block (wave32)
#define TPB   (WPB * 32)

typedef __attribute__((ext_vector_type(2))) float v2f;
typedef __attribute__((ext_vector_type(8))) float v8f;

__device__ __forceinline__ float wsum(float v) {
#pragma unroll
  for (int o = 16; o; o >>= 1) v += __shfl_xor(v, o, 32);
  return v;
}

__global__ __launch_bounds__(TPB) void fused_fwd(
    const int* __restrict__ user1, const int* __restrict__ item1,
    const int* __restrict__ user2, const int* __restrict__ item2,
    const int* __restrict__ u_his, const int* __restrict__ u_pos, const int* __restrict__ u_mask,
    const int* __restrict__ i_his, const int* __restrict__ i_pos, const int* __restrict__ i_mask,
    const float* __restrict__ user1_emb, const float* __restrict__ item1_emb,
    const float* __restrict__ user1_bias, const float* __restrict__ item1_bias,
    const float* __restrict__ u_pos_emb,
    const float* __restrict__ user2_emb, const float* __restrict__ item2_emb,
    const float* __restrict__ user2_bias, const float* __restrict__ item2_bias,
    const float* __restrict__ i_pos_emb,
    const float* __restrict__ attn_u_W, const float* __restrict__ attn_u_b,
    const float* __restrict__ attn_i_W, const float* __restrict__ attn_i_b,
    float* __restrict__ out, float* __restrict__ reg_ws, int nB, int use_atomic)
{
  __shared__ __align__(16) float s_q[WPB][8][DIM];      // async-staged query/output rows
  __shared__ __align__(16) float s_em[WPB][H_PAD][4];   // mask-folded exp(scores): A-operand source
  __shared__ __align__(16) int   s_off[WPB][H_PAD][4];  // row element-offsets: B-operand source
  __shared__ __align__(16) float s_W[2][2 * DIM];       // [0]=attn_u_W, [1]=attn_i_W
  __shared__ float s_reg[WPB][2];

  const int lane = threadIdx.x & 31;
  const int wave = threadIdx.x >> 5;
  const int b    = blockIdx.x * WPB + wave;
  const bool alive = (b < nB);

  for (int t = threadIdx.x; t < 2 * DIM; t += TPB) {
    s_W[0][t] = attn_u_W[t];
    s_W[1][t] = attn_i_W[t];
  }
  if (lane == 0) { s_reg[wave][0] = 0.f; s_reg[wave][1] = 0.f; }

  int u1 = 0, it1 = 0, u2 = 0, it2 = 0;
  if (alive) { u1 = user1[b]; it1 = item1[b]; u2 = user2[b]; it2 = item2[b]; }

  // ---- CDNA5 async DMA: stage the 8 query/output embedding rows into LDS ----
  // Rows: 0 u1s=user1_emb[u1] 1 u1r=user2_emb[u1] 2 i1s=item1_emb[it1] 3 i1r=item2_emb[it1]
  //       4 u2s=user2_emb[u2] 5 u2r=user1_emb[u2] 6 i2s=item2_emb[it2] 7 i2r=item1_emb[it2]
  if (alive) {
    const int hi  = lane >> 4;     // lanes 0-15 -> even row, 16-31 -> odd row
    const int sub = lane & 15;     // 16B chunk within 256B row
    const uint32_t lbase = (uint32_t)(uintptr_t)(&s_q[wave][0][0]) + (uint32_t)sub * 16u;
    uint64_t ga; uint32_t la;
    ga = (uint64_t)(uintptr_t)((hi ? user2_emb : user1_emb) + (size_t)u1 * DIM + sub * 4);
    la = lbase + (uint32_t)(0 + hi) * 256u;
    asm volatile("global_load_async_to_lds_b128 %0, %1, off" :: "v"(la), "v"(ga) : "memory");
    ga = (uint64_t)(uintptr_t)((hi ? item2_emb : item1_emb) + (size_t)it1 * DIM + sub * 4);
    la = lbase + (uint32_t)(2 + hi) * 256u;
    asm volatile("global_load_async_to_lds_b128 %0, %1, off" :: "v"(la), "v"(ga) : "memory");
    ga = (uint64_t)(uintptr_t)((hi ? user1_emb : user2_emb) + (size_t)u2 * DIM + sub * 4);
    la = lbase + (uint32_t)(4 + hi) * 256u;
    asm volatile("global_load_async_to_lds_b128 %0, %1, off" :: "v"(la), "v"(ga) : "memory");
    ga = (uint64_t)(uintptr_t)((hi ? item1_emb : item2_emb) + (size_t)it2 * DIM + sub * 4);
    la = lbase + (uint32_t)(6 + hi) * 256u;
    asm volatile("global_load_async_to_lds_b128 %0, %1, off" :: "v"(la), "v"(ga) : "memory");
  }

  __syncthreads();  // s_W ready

  // ---- Phase 1: per-history key dot-products (lanes over h) ----
  float kd_u[2] = {0.f, 0.f}, kd_i[2] = {0.f, 0.f};
  float mu_r[2] = {0.f, 0.f}, mi_r[2] = {0.f, 0.f};
  float reg1_his = 0.f, reg2_his = 0.f;
  if (alive) {
#pragma unroll
    for (int r = 0; r < 2; ++r) {
      const int h = lane + 32 * r;
      if (h < H_LEN) {
        const int base = b * H_LEN + h;
        const int iu = u_his[base], pu = u_pos[base];
        const int ii = i_his[base], pi = i_pos[base];
        const float mu = (u_mask[base] != 0) ? 1.f : 0.f;
        const float mi = (i_mask[base] != 0) ? 1.f : 0.f;
        mu_r[r] = mu; mi_r[r] = mi;
        *(int4*)&s_off[wave][h][0] = make_int4(iu * DIM, pu * DIM, ii * DIM, pi * DIM);
        const float4* eu = (const float4*)(item1_emb + (size_t)iu * DIM);
        const float4* pu4 = (const float4*)(u_pos_emb + (size_t)pu * DIM);
        const float4* ei = (const float4*)(user2_emb + (size_t)ii * DIM);
        const float4* pi4 = (const float4*)(i_pos_emb + (size_t)pi * DIM);
        const float4* wu = (const float4*)(&s_W[0][DIM]);  // bottom half of attn_u_W
        const float4* wi = (const float4*)(&s_W[1][DIM]);  // bottom half of attn_i_W
        float kdu = 0.f, kdi = 0.f, r1 = 0.f, r2 = 0.f;
#pragma unroll 4
        for (int d4 = 0; d4 < DIM / 4; ++d4) {
          float4 a = eu[d4], p = pu4[d4], w = wu[d4];
          kdu += (a.x + p.x) * w.x + (a.y + p.y) * w.y + (a.z + p.z) * w.z + (a.w + p.w) * w.w;
          r1  += a.x * a.x + a.y * a.y + a.z * a.z + a.w * a.w;
          float4 c = ei[d4], q4 = pi4[d4], v = wi[d4];
          kdi += (c.x + q4.x) * v.x + (c.y + q4.y) * v.y + (c.z + q4.z) * v.z + (c.w + q4.w) * v.w;
          r2  += c.x * c.x + c.y * c.y + c.z * c.z + c.w * c.w;
        }
        kd_u[r] = mu * kdu;  kd_i[r] = mi * kdi;   // masked key => masked key-dot
        reg1_his += mu * r1; reg2_his += mi * r2;  // sum((emb*mask)^2)
      } else if (h < H_PAD) {                      // zero-pad K slots 50,51
        *(float4*)&s_em[wave][h][0]  = make_float4(0.f, 0.f, 0.f, 0.f);
        *(int4*)&s_off[wave][h][0]   = make_int4(0, 0, 0, 0);
      }
    }
  }

  // ---- wait for async-staged queries, compute query-side score scalars ----
  asm volatile("s_wait_asynccnt 0x0" ::: "memory");
  float2 q0 = *(const float2*)&s_q[wave][0][2 * lane];   // u1s
  float2 q1 = *(const float2*)&s_q[wave][1][2 * lane];   // u1r
  float2 q2 = *(const float2*)&s_q[wave][2][2 * lane];   // i1s
  float2 q3 = *(const float2*)&s_q[wave][3][2 * lane];   // i1r
  const float2 WuT = *(const float2*)&s_W[0][2 * lane];  // top half of attn_u_W
  const float2 WiT = *(const float2*)&s_W[1][2 * lane];
  const float qd_us = wsum(q2.x * WuT.x + q2.y * WuT.y);  // i1s . WuTop
  const float qd_ur = wsum(q3.x * WuT.x + q3.y * WuT.y);  // i1r . WuTop
  const float qd_ir = wsum(q1.x * WiT.x + q1.y * WiT.y);  // u1r . WiTop
  const float qd_is = wsum(q0.x * WiT.x + q0.y * WiT.y);  // u1s . WiTop
  const float bu = attn_u_b[0], bi = attn_i_b[0];

  // ---- scores -> exp; raw e feeds denominators, mask-folded e feeds WMMA A ----
  float den[4] = {0.f, 0.f, 0.f, 0.f};
#pragma unroll
  for (int r = 0; r < 2; ++r) {
    const int h = lane + 32 * r;
    if (alive && h < H_LEN) {
      const float e_us = __expf(tanhf(qd_us + kd_u[r] + bu));
      const float e_ur = __expf(tanhf(qd_ur + kd_u[r] + bu));
      const float e_ir = __expf(tanhf(qd_ir + kd_i[r] + bi));
      const float e_is = __expf(tanhf(qd_is + kd_i[r] + bi));
      *(float4*)&s_em[wave][h][0] =
          make_float4(e_us * mu_r[r], e_ur * mu_r[r], e_ir * mi_r[r], e_is * mi_r[r]);
      den[0] += e_us; den[1] += e_ur; den[2] += e_ir; den[3] += e_is;
    }
  }
  den[0] = wsum(den[0]); den[1] = wsum(den[1]);
  den[2] = wsum(den[2]); den[3] = wsum(den[3]);
  const float inv_us = 1.f / den[0], inv_ur = 1.f / den[1];
  const float inv_ir = 1.f / den[2], inv_is = 1.f / den[3];

  // ---- Phase 2: weighted sums via V_WMMA_F32_16X16X4_F32 ----
  // out(2x64) = Em(2x52) @ RawKey(52x64) per key; rows 0/1 = the two query variants.
  // A layout (16x4 f32): lane m(0-15) holds A[m][0..1]; lane m+16 holds A[m][2..3].
  // B layout (4x16 f32): lane n(0-15) holds B[0..1][n]; lane n+16 holds B[2..3][n].
  // C/D: VGPR j, lanes 0-15 = row M=j -> rows 0,1 live in C[0],C[1] on lanes 0-15.
  float p_us = 0.f, p_ur = 0.f, p_ir = 0.f, p_is = 0.f;
  float rq1 = 0.f, rq2 = 0.f;
  if (alive) {
    const int m    = lane & 15;
    const int koff = (lane >> 4) * 2;
#pragma unroll
    for (int t = 0; t < 4; ++t) {           // N tiles over D=64
      v8f Cu = {}; v8f Ci = {};
      const int n = 16 * t + m;
      for (int c = 0; c < 13; ++c) {        // K chunks over H_PAD=52
        const int ha = 4 * c + koff, hb = ha + 1;
        const float4 ea = *(const float4*)&s_em[wave][ha][0];
        const float4 eb = *(const float4*)&s_em[wave][hb][0];
        const int4 oa = *(const int4*)&s_off[wave][ha][0];
        const int4 ob = *(const int4*)&s_off[wave][hb][0];
        v2f Au, Ai, Bu, Bi;
        Au[0] = (m == 0) ? ea.x : ((m == 1) ? ea.y : 0.f);
        Au[1] = (m == 0) ? eb.x : ((m == 1) ? eb.y : 0.f);
        Ai[0] = (m == 0) ? ea.z : ((m == 1) ? ea.w : 0.f);
        Ai[1] = (m == 0) ? eb.z : ((m == 1) ? eb.w : 0.f);
        Bu[0] = item1_emb[oa.x + n] + u_pos_emb[oa.y + n];
        Bu[1] = item1_emb[ob.x + n] + u_pos_emb[ob.y + n];
        Bi[0] = user2_emb[oa.z + n] + i_pos_emb[oa.w + n];
        Bi[1] = user2_emb[ob.z + n] + i_pos_emb[ob.w + n];
        Cu = __builtin_amdgcn_wmma_f32_16x16x4_f32(false, Au, false, Bu, (short)0, Cu, false, false);
        Ci = __builtin_amdgcn_wmma_f32_16x16x4_f32(false, Ai, false, Bi, (short)0, Ci, false, false);
      }
      // extract rows 0,1 and fold into the logit / regularizer partial dots
      const float u_ds = Cu[0] * inv_us, u_dr = Cu[1] * inv_ur;
      const float i_dr = Ci[0] * inv_ir, i_ds = Ci[1] * inv_is;
      const int d = 16 * t + m;
      const float u1s_d = s_q[wave][0][d], u1r_d = s_q[wave][1][d];
      const float i1s_d = s_q[wave][2][d], i1r_d = s_q[wave][3][d];
      const float u2s_d = s_q[wave][4][d], u2r_d = s_q[wave][5][d];
      const float i2s_d = s_q[wave][6][d], i2r_d = s_q[wave][7][d];
      if (lane < 16) {
        p_us += (u_ds + i1s_d) * u1s_d;
        p_ur += (u_dr + i2r_d) * u2r_d;
        p_ir += (i_dr + i1r_d) * u1r_d;
        p_is += (i_ds + i2s_d) * u2s_d;
        rq1  += u1s_d * u1s_d + i1s_d * i1s_d;
        rq2  += u2s_d * u2s_d + i2s_d * i2s_d;
      }
    }
  }

  p_us = wsum(p_us); p_ur = wsum(p_ur); p_ir = wsum(p_ir); p_is = wsum(p_is);
  const float r1 = wsum(rq1 + reg1_his);
  const float r2 = wsum(rq2 + reg2_his);

  if (alive && lane == 0) {
    out[b]          = user1_bias[u1] + item1_bias[it1] + p_us;   // logits_us
    out[nB + b]     = user2_bias[u1] + item2_bias[it1] + p_ir;   // logits_ir
    out[2 * nB + b] = user1_bias[u2] + item1_bias[it2] + p_ur;   // logits_ur
    out[3 * nB + b] = user2_bias[u2] + item2_bias[it2] + p_is;   // logits_is
    s_reg[wave][0] = r1; s_reg[wave][1] = r2;
  }
  __syncthreads();
  if (threadIdx.x == 0) {
    float a = 0.f, c = 0.f;
#pragma unroll
    for (int w = 0; w < WPB; ++w) { a += s_reg[w][0]; c += s_reg[w][1]; }
    if (use_atomic) {
      atomicAdd(&out[4 * nB], a);
      atomicAdd(&out[4 * nB + 1], c);
    } else {
      reg_ws[2 * blockIdx.x]     = a;
      reg_ws[2 * blockIdx.x + 1] = c;
    }
  }
}

__global__ void reduce_regs(const float* __restrict__ reg_ws, float* __restrict__ out,
                            int nblocks, int nB) {
  __shared__ float sh[2][256];
  float a = 0.f, c = 0.f;
  for (int i = threadIdx.x; i < nblocks; i += 256) { a += reg_ws[2 * i]; c += reg_ws[2 * i + 1]; }
  sh[0][threadIdx.x] = a; sh[1][threadIdx.x] = c;
  __syncthreads();
  for (int s = 128; s; s >>= 1) {
    if (threadIdx.x < (unsigned)s) {
      sh[0][threadIdx.x] += sh[0][threadIdx.x + s];
      sh[1][threadIdx.x] += sh[1][threadIdx.x + s];
    }
    __syncthreads();
  }
  if (threadIdx.x == 0) { out[4 * nB] = sh[0][0]; out[4 * nB + 1] = sh[1][0]; }
}

extern "C" void kernel_launch(void* const* d_in, const int* in_sizes, int n_in,
                              void* d_out, int out_size, void* d_ws, size_t ws_size,
                              hipStream_t stream) {
  const int nB = in_sizes[0];
  const int nblocks = (nB + WPB - 1) / WPB;
  const bool use_ws = ws_size >= (size_t)nblocks * 2 * sizeof(float);
  float* out = (float*)d_out;
  if (!use_ws) {
    hipMemsetAsync((char*)d_out + (size_t)4 * nB * sizeof(float), 0, 2 * sizeof(float), stream);
  }
  fused_fwd<<<nblocks, TPB, 0, stream>>>(
      (const int*)d_in[0], (const int*)d_in[1], (const int*)d_in[2], (const int*)d_in[3],
      (const int*)d_in[4], (const int*)d_in[5], (const int*)d_in[6],
      (const int*)d_in[7], (const int*)d_in[8], (const int*)d_in[9],
      (const float*)d_in[10], (const float*)d_in[11], (const float*)d_in[12], (const float*)d_in[13],
      (const float*)d_in[14],
      (const float*)d_in[15], (const float*)d_in[16], (const float*)d_in[17], (const float*)d_in[18],
      (const float*)d_in[19],
      (const float*)d_in[20], (const float*)d_in[21], (const float*)d_in[22], (const float*)d_in[23],
      out, (float*)d_ws, nB, use_ws ? 0 : 1);
  if (use_ws) {
    reduce_regs<<<1, 256, 0, stream>>>((const float*)d_ws, out, nblocks, nB);
  }
}